// CensusLoss_33809982554508
// MI455X (gfx1250) — compile-verified
//
#include <hip/hip_runtime.h>
#include <cstdint>

// Problem constants (from reference)
#define RAD   4
#define WIN   9
#define MAXD  192
#define B_    2
#define H_    256
#define W_    512
#define HC    248     // H - 2*RAD
#define WC    504     // W - 2*RAD

// Tiling: 32x8 outputs per block (256 threads = 8 wave32)
#define TX    32
#define TY    8
#define TILH  16      // TY + 2*RAD
#define LTW   40      // TX + 2*RAD               (left tile width, floats)
#define RTW   232     // TX + (MAXD-1) + WIN      (right tile width, floats)

typedef unsigned u32x4 __attribute__((ext_vector_type(4)));
typedef unsigned u32x8 __attribute__((ext_vector_type(8)));

// per-lane async global->LDS 16-byte copy (gfx1250 async engine, ASYNCcnt)
__device__ __forceinline__ void async_copy_b128(unsigned lds_byte_off, const float* gaddr) {
    asm volatile("global_load_async_to_lds_b128 %0, %1, off"
                 :: "v"(lds_byte_off), "v"(gaddr)
                 : "memory");
}

__device__ __forceinline__ void wait_async_done() {
    asm volatile("s_wait_asynccnt 0" ::: "memory");
}

// TDM 2D tile load: one instruction moves a tileW x TILH tile (row stride W_)
// from global to LDS; OOB columns (>= tensor_dim0) are zero-filled by hardware.
// D# layout per CDNA5 ISA (2D: groups 0 and 1 only).
__device__ __forceinline__ void tdm_load_tile_2d(unsigned lds_off,
                                                 const float* gtile,
                                                 unsigned tensor_d0,
                                                 unsigned tensor_d1,
                                                 unsigned tileW) {
    const unsigned long long ga = (unsigned long long)(uintptr_t)gtile;
    u32x4 g0;
    g0.x = 1u;                                                  // count=1 (valid), user mode
    g0.y = lds_off;                                             // LDS byte address
    g0.z = (unsigned)(ga & 0xFFFFFFFFull);                      // global_addr[31:0]
    g0.w = ((unsigned)(ga >> 32) & 0x01FFFFFFu) | 0x80000000u;  // global_addr[56:32] | type=2
    u32x8 g1;
    g1.s0 = 0x00020000u;                                        // data_size=2 (4 bytes)
    g1.s1 = (tensor_d0 & 0xFFFFu) << 16;                        // tensor_dim0[15:0]
    g1.s2 = (tensor_d0 >> 16) | ((tensor_d1 & 0xFFFFu) << 16);  // tensor_dim0[31:16] | tensor_dim1[15:0]
    g1.s3 = (tensor_d1 >> 16) | (tileW << 16);                  // tensor_dim1[31:16] | tile_dim0
    g1.s4 = (unsigned)TILH;                                     // tile_dim1=16, tile_dim2=0 (2D)
    g1.s5 = (unsigned)W_;                                       // tensor_dim0_stride[31:0] = 512
    g1.s6 = 0u;                                                 // stride[47:32] | tensor_dim1_stride lo
    g1.s7 = 0u;                                                 // tensor_dim1_stride hi
    asm volatile("tensor_load_to_lds %0, %1"
                 :: "s"(g0), "s"(g1)
                 : "memory");
}

__global__ void __launch_bounds__(256)
census_loss_init(unsigned* __restrict__ accum) {
    if (threadIdx.x < 2) accum[threadIdx.x] = 0u;
}

__global__ void __launch_bounds__(256)
census_loss_fused(const float* __restrict__ left,
                  const float* __restrict__ right,
                  const float* __restrict__ disp,
                  unsigned* __restrict__ accum) {
    __shared__ __align__(16) float tl[TILH * LTW];
    __shared__ __align__(16) float tr[TILH * RTW];

    const int tx  = threadIdx.x;            // 0..31 (one wave per row)
    const int ty  = threadIdx.y;            // 0..7
    const int tid = ty * TX + tx;           // 0..255
    const int x0  = blockIdx.x * TX;
    const int y0  = blockIdx.y * TY;
    const int b   = blockIdx.z;

    const float* limg = left  + (size_t)b * H_ * W_;
    const float* rimg = right + (size_t)b * H_ * W_;

    const unsigned tl_base = (unsigned)(uintptr_t)(void*)tl;
    const unsigned tr_base = (unsigned)(uintptr_t)(void*)tr;

    // ---- right tile (16 x 232 = 14.5 KB): single TDM DMA, wave 0 only ----
    // tensor_dim0 = W - x0 => columns past the image row are hardware zero-filled
    if (ty == 0) {
        tdm_load_tile_2d(tr_base,
                         rimg + (size_t)y0 * W_ + x0,
                         (unsigned)(W_ - x0), (unsigned)(H_ - y0),
                         (unsigned)RTW);
    }

    // ---- left tile (16 x 40): per-lane async b128 copies, waves 0-4 (160 chunks) ----
    for (int c = tid; c < (TILH * LTW / 4); c += 256) {
        int row = c / (LTW / 4);
        int col = (c % (LTW / 4)) * 4;
        int gx  = x0 + col; if (gx > W_ - 4) gx = W_ - 4;   // clamped chunks are never read
        async_copy_b128(tl_base + (unsigned)c * 16u,
                        limg + (size_t)(y0 + row) * W_ + gx);
    }
    wait_async_done();                       // this wave's async LDS writes complete
    if (ty == 0) __builtin_amdgcn_s_wait_tensorcnt(0);  // TDM tile complete (issuing wave)
    __syncthreads();                         // all waves see both tiles

    const int h   = y0 + ty;        // 0..247 (31*8 == 248, always in crop)
    const int w   = x0 + tx;        // 0..511
    const bool inb = (w < WC);

    // ---- left census: pack 80 (win > center) bits into 64+16 ----
    const float lc0 = tl[(ty + RAD) * LTW + tx + RAD];
    unsigned long long llo = 0ull; unsigned lhi = 0u;
    {
        int k = 0;
        #pragma unroll
        for (int i = 0; i < WIN; ++i) {
            #pragma unroll
            for (int j = 0; j < WIN; ++j) {
                if (i == RAD && j == RAD) continue;
                unsigned bit = (tl[(ty + i) * LTW + tx + j] > lc0) ? 1u : 0u;
                if (k < 64) llo |= (unsigned long long)bit << k;
                else        lhi |= bit << (k - 64);
                ++k;
            }
        }
    }

    unsigned cost = 0u, cnt = 0u;
    if (inb) {
        const float dsp = disp[(size_t)(b * H_ + h + RAD) * W_ + (w + RAD)];
        if (dsp > 0.0f) {
            cnt = 1u;
            int d = (int)dsp;
            d = d < 0 ? 0 : (d > MAXD - 1 ? MAXD - 1 : d);
            if (w + d < WC) {
                // right census at crop col w+d  ->  tile col tx+d (<= 222; +8 < 232)
                const int cb = tx + d;
                const float rc0 = tr[(ty + RAD) * RTW + cb + RAD];
                unsigned long long rlo = 0ull; unsigned rhi = 0u;
                int k = 0;
                #pragma unroll
                for (int i = 0; i < WIN; ++i) {
                    #pragma unroll
                    for (int j = 0; j < WIN; ++j) {
                        if (i == RAD && j == RAD) continue;
                        unsigned bit = (tr[(ty + i) * RTW + cb + j] > rc0) ? 1u : 0u;
                        if (k < 64) rlo |= (unsigned long long)bit << k;
                        else        rhi |= bit << (k - 64);
                        ++k;
                    }
                }
                cost = (unsigned)__popcll(llo ^ rlo) + (unsigned)__popc(lhi ^ rhi);
            } else {
                // invalid shift: right census channels are all zero in reference
                cost = (unsigned)__popcll(llo) + (unsigned)__popc(lhi);
            }
        }
    }

    // ---- wave32 tree reduction (integer => exact & order-independent) ----
    #pragma unroll
    for (int off = 16; off > 0; off >>= 1) {
        cost += __shfl_down(cost, off, 32);
        cnt  += __shfl_down(cnt,  off, 32);
    }
    if (tx == 0) {  // blockDim.x == 32 => each block row is one wave32
        atomicAdd(&accum[0], cost);
        atomicAdd(&accum[1], cnt);
    }
}

__global__ void census_loss_finalize(const unsigned* __restrict__ accum,
                                     float* __restrict__ out) {
    out[0] = (float)accum[0] / ((float)accum[1] + 1e-6f);
}

extern "C" void kernel_launch(void* const* d_in, const int* in_sizes, int n_in,
                              void* d_out, int out_size, void* d_ws, size_t ws_size,
                              hipStream_t stream) {
    const float* left  = (const float*)d_in[0];   // (B,1,H,W) f32
    const float* right = (const float*)d_in[1];   // (B,1,H,W) f32
    const float* disp  = (const float*)d_in[2];   // (B,1,H,W) f32
    unsigned* accum = (unsigned*)d_ws;            // [cost_sum, mask_count]
    float* out = (float*)d_out;                   // scalar

    census_loss_init<<<dim3(1), dim3(32), 0, stream>>>(accum);

    dim3 block(TX, TY);
    dim3 grid((WC + TX - 1) / TX,   // 16
              (HC + TY - 1) / TY,   // 31
              B_);                  // 2
    census_loss_fused<<<grid, block, 0, stream>>>(left, right, disp, accum);

    census_loss_finalize<<<dim3(1), dim3(1), 0, stream>>>(accum, out);
}